// MZILayer_18837726561084
// MI455X (gfx1250) — compile-verified
//
#include <hip/hip_runtime.h>
#include <math.h>

#define NN 256
#define BATCH 4096
#define N_MZIS 32640
#define LAYERS 255

typedef __attribute__((ext_vector_type(2))) float v2f;
typedef __attribute__((ext_vector_type(8))) float v8f;

// ---------------------------------------------------------------------------
// Kernel 1: precompute per-MZI coefficients (wave-uniform in the mesh loop).
// coef[i] = { cos(theta), sin(theta)*cos(phi), sin(theta)*sin(phi), 0 }
// ---------------------------------------------------------------------------
__global__ void __launch_bounds__(256)
mzi_coef_kernel(const float* __restrict__ theta,
                const float* __restrict__ phi,
                float4* __restrict__ coef) {
    int i = blockIdx.x * blockDim.x + threadIdx.x;
    if (i < N_MZIS) {
        float s, c, sp, cp;
        sincosf(theta[i], &s, &c);
        sincosf(phi[i], &sp, &cp);
        coef[i] = make_float4(c, s * cp, s * sp, 0.0f);
    }
}

// ---------------------------------------------------------------------------
// Kernel 2: build W = Re(diag(e^{i phi_ext}) * U) for a 32-column slice.
// Each thread owns ONE column of U (columns evolve independently under the
// MZI scan), kept in LDS: 256 rows x 32 cols x (re,im) = 64 KB. The 32640-step
// scan needs no barriers. Coefficient reads are loop-uniform -> scalar loads.
// ---------------------------------------------------------------------------
__global__ void __launch_bounds__(32)
mzi_mesh_kernel(const float4* __restrict__ coef,
                const float* __restrict__ phi_ext,
                float* __restrict__ W) {
    __shared__ float Ure[NN][32];
    __shared__ float Uim[NN][32];
    const int t   = threadIdx.x;            // 0..31, one column per lane
    const int col = blockIdx.x * 32 + t;

    // U0 = identity (this column slice)
    for (int r = 0; r < NN; ++r) {
        Ure[r][t] = (r == col) ? 1.0f : 0.0f;
        Uim[r][t] = 0.0f;
    }

    // Reck schedule: layer l has positions 0..(254-l); strict scan order.
    int idx = 0;
    for (int layer = 0; layer < LAYERS; ++layer) {
        const int npos = LAYERS - layer;
        for (int pos = 0; pos < npos; ++pos, ++idx) {
            const float4 cf = coef[idx];    // uniform -> s_load_b128
            const float c  = cf.x;          // cos(theta)
            const float er = cf.y;          // sin(theta)*cos(phi)
            const float ei = cf.z;          // sin(theta)*sin(phi)
            const float r0r = Ure[pos    ][t], r0i = Uim[pos    ][t];
            const float r1r = Ure[pos + 1][t], r1i = Uim[pos + 1][t];
            // new0 = c*r0 - s*conj(e)*r1 ;  new1 = s*e*r0 + c*r1
            const float n0r = fmaf(c, r0r, -fmaf(er, r1r,  ei * r1i));
            const float n0i = fmaf(c, r0i, -fmaf(er, r1i, -ei * r1r));
            const float n1r = fmaf(c, r1r,  fmaf(er, r0r, -ei * r0i));
            const float n1i = fmaf(c, r1i,  fmaf(er, r0i,  ei * r0r));
            Ure[pos    ][t] = n0r; Uim[pos    ][t] = n0i;
            Ure[pos + 1][t] = n1r; Uim[pos + 1][t] = n1i;
        }
    }

    // Phase screen + take real part: W[j][k] = cos(pe_j)*Re - sin(pe_j)*Im.
    // Row-major W so the GEMM B-fragment loads are contiguous b64s.
    for (int j = 0; j < NN; ++j) {
        float sj, cj;
        sincosf(phi_ext[j], &sj, &cj);      // uniform
        W[j * NN + col] = fmaf(cj, Ure[j][t], -sj * Uim[j][t]);
    }
}

// ---------------------------------------------------------------------------
// Kernel 3: out[b][j] = sum_k x[b][k] * W[j][k]   (out = x * W^T, fp32 GEMM)
// via V_WMMA_F32_16X16X4_F32. Each wave owns a 32x64 output super-tile:
//   2 A-frags (rows m, m+16 of x) x 4 B-frags (4 adjacent N-tiles of W^T)
//   -> 8 INDEPENDENT wmma accumulation chains per K-step (hides XDL latency)
//   -> ~10.7 flops per L2 byte vs ~4 for a single 16x16 tile
// A-frag (16x4): lane lo = row, hi selects {K,K+1} vs {K+2,K+3} -> v2f load.
// B-frag (4x16): B[kk][n] = W[n][kk] -> contiguous v2f from W's row n.
// EXEC is all-ones (full waves, no divergence) as WMMA requires.
// ---------------------------------------------------------------------------
__global__ void __launch_bounds__(256)
mzi_gemm_kernel(const float* __restrict__ x,
                const float* __restrict__ W,
                float* __restrict__ out) {
    const int lane  = threadIdx.x & 31;
    const int wave  = threadIdx.x >> 5;
    const int tile  = blockIdx.x * 8 + wave;   // 512 super-tiles total
    const int tM    = tile >> 2;               // 0..127  (BATCH/32)
    const int tN    = tile & 3;                // 0..3    (NN/64)
    const int lo    = lane & 15;
    const int hi    = lane >> 4;               // 0 -> K=0,1 ; 1 -> K=2,3

    const float* aptr0 = x + (tM * 32      + lo) * NN + hi * 2;
    const float* aptr1 = aptr0 + 16 * NN;
    const float* bptr0 = W + (tN * 64      + lo) * NN + hi * 2;
    const float* bptr1 = bptr0 + 16 * NN;
    const float* bptr2 = bptr0 + 32 * NN;
    const float* bptr3 = bptr0 + 48 * NN;

    v8f c00 = {}, c01 = {}, c02 = {}, c03 = {};
    v8f c10 = {}, c11 = {}, c12 = {}, c13 = {};

#pragma unroll 2
    for (int k = 0; k < NN; k += 4) {
        v2f a0 = *(const v2f*)(aptr0 + k);
        v2f a1 = *(const v2f*)(aptr1 + k);
        v2f b0 = *(const v2f*)(bptr0 + k);
        v2f b1 = *(const v2f*)(bptr1 + k);
        v2f b2 = *(const v2f*)(bptr2 + k);
        v2f b3 = *(const v2f*)(bptr3 + k);
        c00 = __builtin_amdgcn_wmma_f32_16x16x4_f32(false, a0, false, b0, (short)0, c00, false, false);
        c01 = __builtin_amdgcn_wmma_f32_16x16x4_f32(false, a0, false, b1, (short)0, c01, false, false);
        c02 = __builtin_amdgcn_wmma_f32_16x16x4_f32(false, a0, false, b2, (short)0, c02, false, false);
        c03 = __builtin_amdgcn_wmma_f32_16x16x4_f32(false, a0, false, b3, (short)0, c03, false, false);
        c10 = __builtin_amdgcn_wmma_f32_16x16x4_f32(false, a1, false, b0, (short)0, c10, false, false);
        c11 = __builtin_amdgcn_wmma_f32_16x16x4_f32(false, a1, false, b1, (short)0, c11, false, false);
        c12 = __builtin_amdgcn_wmma_f32_16x16x4_f32(false, a1, false, b2, (short)0, c12, false, false);
        c13 = __builtin_amdgcn_wmma_f32_16x16x4_f32(false, a1, false, b3, (short)0, c13, false, false);
    }

    // D layout: VGPR g -> (M = g + 8*hi, N = lo) within each 16x16 tile.
    v8f acc[2][4] = {{c00, c01, c02, c03}, {c10, c11, c12, c13}};
#pragma unroll
    for (int i = 0; i < 2; ++i) {
#pragma unroll
        for (int j = 0; j < 4; ++j) {
            float* optr = out + (tM * 32 + i * 16 + hi * 8) * NN
                              + tN * 64 + j * 16 + lo;
#pragma unroll
            for (int g = 0; g < 8; ++g)
                optr[g * NN] = acc[i][j][g];
        }
    }
}

// ---------------------------------------------------------------------------
extern "C" void kernel_launch(void* const* d_in, const int* in_sizes, int n_in,
                              void* d_out, int out_size, void* d_ws, size_t ws_size,
                              hipStream_t stream) {
    const float* x       = (const float*)d_in[0];   // [4096*256]
    const float* theta   = (const float*)d_in[1];   // [32640]
    const float* phi_int = (const float*)d_in[2];   // [32640]
    const float* phi_ext = (const float*)d_in[3];   // [256]
    float*       out     = (float*)d_out;           // [4096*256]

    float4* coef = (float4*)d_ws;                                   // 510 KB
    float*  W    = (float*)((char*)d_ws + (size_t)N_MZIS * 16);     // 256 KB

    mzi_coef_kernel<<<(N_MZIS + 255) / 256, 256, 0, stream>>>(theta, phi_int, coef);
    mzi_mesh_kernel<<<NN / 32, 32, 0, stream>>>(coef, phi_ext, W);
    mzi_gemm_kernel<<<(BATCH / 32) * (NN / 64) / 8, 256, 0, stream>>>(x, W, out);
}